// EMOGI_59528246723156
// MI455X (gfx1250) — compile-verified
//
#include <hip/hip_runtime.h>

typedef float v2f __attribute__((ext_vector_type(2)));
typedef float v8f __attribute__((ext_vector_type(8)));

#define NF   1024   // F*C = 64*16 (reduction length of layer-1 GEMM)
#define H1   20
#define H1P  32     // padded to two 16-wide WMMA column tiles
#define H2   40
#define NSTEP (NF / 4)   // 256 WMMA k-steps (K=4 per v_wmma_f32_16x16x4_f32)

// ---------------------------------------------------------------------------
// Pack w1 [F=64][H1=20][C=16] into WMMA-B-ready layout:
//   Bw[step][tile][lane][pair], pair within a lane = consecutive VGPRs.
// ISA B layout (4x16 f32): VGPR0 = rows K=0 (lanes0-15) / K=2 (lanes16-31),
//                          VGPR1 = rows K=1 / K=3; col = lane%16.
// B[m][k] with m = f*16 + c equals w1[f][k][c]; cols 20..31 are zero.
// ---------------------------------------------------------------------------
__global__ __launch_bounds__(256) void pack_w1_kernel(const float* __restrict__ w1,
                                                      float* __restrict__ Bw) {
    int idx = blockIdx.x * blockDim.x + threadIdx.x;   // 256*2*32*2 = 32768 total
    if (idx >= NSTEP * 2 * 32 * 2) return;
    int pair = idx & 1;
    int lane = (idx >> 1) & 31;
    int tile = (idx >> 6) & 1;
    int step = idx >> 7;
    int kk = 2 * (lane >> 4) + pair;          // K index within step, 0..3
    int m  = step * 4 + kk;                   // 0..1023
    int k  = tile * 16 + (lane & 15);         // output column 0..31
    int f = m >> 4, c = m & 15;
    float v = (k < H1) ? w1[(f * H1 + k) * 16 + c] : 0.0f;
    Bw[idx] = v;
}

// ---------------------------------------------------------------------------
// Layer-1 GEMM: pre[n,k] = sum_m x[n,m] * B[m,k]  via V_WMMA_F32_16X16X4_F32.
// One wave per 16-row node tile; two column tiles accumulated in v8f each.
// A layout (16x4 f32): lane L holds row L%16, VGPR0=K(2*(L/16)), VGPR1=+1.
// ---------------------------------------------------------------------------
__global__ __launch_bounds__(256) void gemm1_wmma_kernel(const float* __restrict__ x,
                                                         const float* __restrict__ Bw,
                                                         float* __restrict__ pre,
                                                         int n, int nTiles) {
    int wave = threadIdx.x >> 5;
    int lane = threadIdx.x & 31;
    int tileIdx = blockIdx.x * 8 + wave;
    if (tileIdx >= nTiles) return;            // uniform per wave: EXEC stays all-1s
    int rowBase = tileIdx * 16;
    int r    = lane & 15;
    int half = lane >> 4;
    int rsrc = rowBase + r;
    if (rsrc >= n) rsrc = n - 1;              // clamp (N=100000 is 16-aligned anyway)
    const float* xrow = x + (size_t)rsrc * NF + 2 * half;
    const float* bp   = Bw + (size_t)lane * 2;

    v8f c0 = {};
    v8f c1 = {};
    for (int step = 0; step < NSTEP; ++step) {
        v2f a  = *(const v2f*)(xrow + step * 4);
        v2f b0 = *(const v2f*)(bp + (size_t)(step * 2 + 0) * 64);
        v2f b1 = *(const v2f*)(bp + (size_t)(step * 2 + 1) * 64);
        c0 = __builtin_amdgcn_wmma_f32_16x16x4_f32(false, a, false, b0,
                                                   (short)0, c0, false, false);
        c1 = __builtin_amdgcn_wmma_f32_16x16x4_f32(false, a, false, b1,
                                                   (short)0, c1, false, false);
    }

    // C/D layout: VGPR rr of lane L -> row rr + 8*(L/16), col L%16
    int col = lane & 15;
    #pragma unroll
    for (int rr = 0; rr < 8; ++rr) {
        int M = rr + 8 * half;
        int nrow = rowBase + M;
        if (nrow < n) {
            pre[(size_t)nrow * H1P + col]      = c0[rr];
            pre[(size_t)nrow * H1P + 16 + col] = c1[rr];
        }
    }
}

// ---------------------------------------------------------------------------
// COO SpMM: out[row[e], k] += val[e] * h[col[e], k]   (one thread per (e,k))
// K constexpr -> fast division; LDH = input row stride (pre is padded to 32).
// ---------------------------------------------------------------------------
template <int K, int LDH>
__global__ __launch_bounds__(256) void spmm_kernel(const int* __restrict__ row,
                                                   const int* __restrict__ col,
                                                   const float* __restrict__ val,
                                                   const float* __restrict__ h,
                                                   float* __restrict__ out,
                                                   long long total) {
    long long idx = (long long)blockIdx.x * blockDim.x + threadIdx.x;
    if (idx >= total) return;
    int e, k;
    if (K == 1) { e = (int)idx; k = 0; }
    else        { e = (int)(idx / K); k = (int)(idx - (long long)e * K); }
    int c = col[e];
    int rdst = row[e];
    float v = val[e] * h[(size_t)c * LDH + k];
    unsafeAtomicAdd(&out[(size_t)rdst * K + k], v);
}

// ---------------------------------------------------------------------------
// Layer-2 dense: t2 = relu(h1) @ w2   (w2 20x40 staged in LDS, thread per node)
// ---------------------------------------------------------------------------
__global__ __launch_bounds__(256) void gemm2_kernel(const float* __restrict__ h1,
                                                    const float* __restrict__ w2,
                                                    float* __restrict__ t2, int n) {
    __shared__ float w2s[H1 * H2];
    for (int i = threadIdx.x; i < H1 * H2; i += blockDim.x) w2s[i] = w2[i];
    __syncthreads();
    int nid = blockIdx.x * blockDim.x + threadIdx.x;
    if (nid >= n) return;
    float r[H1];
    #pragma unroll
    for (int k = 0; k < H1; ++k) r[k] = fmaxf(h1[(size_t)nid * H1 + k], 0.0f);
    #pragma unroll 4
    for (int j = 0; j < H2; ++j) {
        float acc = 0.0f;
        #pragma unroll
        for (int k = 0; k < H1; ++k) acc = fmaf(r[k], w2s[k * H2 + j], acc);
        t2[(size_t)nid * H2 + j] = acc;
    }
}

// ---------------------------------------------------------------------------
// Output dense: t3 = relu(h2) @ w3   (w3 is 40x1)
// ---------------------------------------------------------------------------
__global__ __launch_bounds__(256) void gemm3_kernel(const float* __restrict__ h2,
                                                    const float* __restrict__ w3,
                                                    float* __restrict__ t3, int n) {
    int nid = blockIdx.x * blockDim.x + threadIdx.x;
    if (nid >= n) return;
    float acc = 0.0f;
    #pragma unroll
    for (int j = 0; j < H2; ++j)
        acc = fmaf(fmaxf(h2[(size_t)nid * H2 + j], 0.0f), w3[j], acc);
    t3[nid] = acc;
}

// ---------------------------------------------------------------------------
// Host-side orchestration (graph-capture safe: kernels + hipMemsetAsync only)
// ---------------------------------------------------------------------------
extern "C" void kernel_launch(void* const* d_in, const int* in_sizes, int n_in,
                              void* d_out, int out_size, void* d_ws, size_t ws_size,
                              hipStream_t stream) {
    const float* x   = (const float*)d_in[0];
    const int*   row = (const int*)  d_in[1];
    const int*   col = (const int*)  d_in[2];
    const float* val = (const float*)d_in[3];
    const float* w1  = (const float*)d_in[4];
    const float* w2  = (const float*)d_in[5];
    const float* w3  = (const float*)d_in[6];
    float* out = (float*)d_out;

    const int N = in_sizes[0] / NF;   // x is [N, F, C] = [N, 1024]
    const int E = in_sizes[1];

    // Workspace partition (256B aligned)
    auto align256 = [](size_t v) { return (v + 255) & ~(size_t)255; };
    char* ws = (char*)d_ws;
    size_t off = 0;
    float* Bw  = (float*)(ws + off); off = align256(off + (size_t)NSTEP * 2 * 32 * 2 * 4);
    float* pre = (float*)(ws + off); off = align256(off + (size_t)N * H1P * 4);
    float* h1  = (float*)(ws + off); off = align256(off + (size_t)N * H1 * 4);
    float* t2  = (float*)(ws + off); off = align256(off + (size_t)N * H2 * 4);
    float* h2  = (float*)(ws + off); off = align256(off + (size_t)N * H2 * 4);
    float* t3  = (float*)(ws + off); off = align256(off + (size_t)N * 4);

    // Atomic accumulation targets must be zeroed every call
    hipMemsetAsync(h1, 0, (size_t)N * H1 * 4, stream);
    hipMemsetAsync(h2, 0, (size_t)N * H2 * 4, stream);
    hipMemsetAsync(out, 0, (size_t)out_size * 4, stream);

    // 0) Pack w1 into WMMA-B layout
    pack_w1_kernel<<<(NSTEP * 2 * 32 * 2 + 255) / 256, 256, 0, stream>>>(w1, Bw);

    // 1) pre = x @ B   (fp32 WMMA)
    int nTiles = (N + 15) / 16;
    gemm1_wmma_kernel<<<(nTiles + 7) / 8, 256, 0, stream>>>(x, Bw, pre, N, nTiles);

    // 2) h1 = A @ pre  (relu deferred to gemm2 read)
    long long tot1 = (long long)E * H1;
    spmm_kernel<H1, H1P><<<(unsigned)((tot1 + 255) / 256), 256, 0, stream>>>(
        row, col, val, pre, h1, tot1);

    // 3) t2 = relu(h1) @ w2
    gemm2_kernel<<<(N + 255) / 256, 256, 0, stream>>>(h1, w2, t2, N);

    // 4) h2 = A @ t2
    long long tot2 = (long long)E * H2;
    spmm_kernel<H2, H2><<<(unsigned)((tot2 + 255) / 256), 256, 0, stream>>>(
        row, col, val, t2, h2, tot2);

    // 5) t3 = relu(h2) @ w3
    gemm3_kernel<<<(N + 255) / 256, 256, 0, stream>>>(h2, w3, t3, N);

    // 6) out = A @ t3
    spmm_kernel<1, 1><<<(unsigned)((E + 255) / 256), 256, 0, stream>>>(
        row, col, val, t3, out, (long long)E);
}